// SCW_27693949124723
// MI455X (gfx1250) — compile-verified
//
#include <hip/hip_runtime.h>

// ---------------------------------------------------------------------------
// SCW coincidence-window detector for MI455X (gfx1250, wave32).
// Memory-bound: 128 MiB read + 128 MiB write  ->  ~11 us at 23.3 TB/s.
// CDNA5 paths used: async global->LDS copies (ASYNCcnt) for a double-buffered
// streaming pipeline, split wait counters, LDS atomics for event compaction.
// ---------------------------------------------------------------------------

#ifndef __has_builtin
#define __has_builtin(x) 0
#endif

#if __has_builtin(__builtin_amdgcn_global_load_async_to_lds_b128)
#define SCW_ASYNC_LDS 1
#else
#define SCW_ASYNC_LDS 0
#endif

#define SCW_AS1 __attribute__((address_space(1)))
#define SCW_AS3 __attribute__((address_space(3)))

typedef int scw_v4i __attribute__((vector_size(16)));   // int4 payload of b128

#if SCW_ASYNC_LDS
// per-wave wait on ASYNCcnt (async loads complete in order within a wave)
#define SCW_WAIT_ASYNC(n) asm volatile("s_wait_asynccnt " #n ::: "memory")
#else
#define SCW_WAIT_ASYNC(n)
#endif

namespace {
constexpr int kT       = 8192;   // time steps
constexpr int kC       = 64;     // channels
constexpr int kWindow  = 8;      // coincidence window
constexpr int kThreads = 1024;   // 32 waves (wave32)
constexpr int kChunk   = 64;     // time rows staged per pipeline step
constexpr int kNChunks = kT / kChunk;             // 128
constexpr int kMaxEv   = 2048;   // expected ~262 events/sample (sigma~16)
}

// ---------------------------------------------------------------------------
// Kernel 1: zero the whole output (128 MiB + trailing scalar).
// ---------------------------------------------------------------------------
__global__ __launch_bounds__(256) void SCW_zero_kernel(float* __restrict__ out,
                                                       size_t n) {
  const size_t stride = (size_t)gridDim.x * blockDim.x;
  size_t i = (size_t)blockIdx.x * blockDim.x + threadIdx.x;
  const size_t n4 = n >> 2;
  float4 z;
  z.x = 0.f; z.y = 0.f; z.z = 0.f; z.w = 0.f;
  for (size_t k = i; k < n4; k += stride)
    reinterpret_cast<float4*>(out)[k] = z;
  // tail (out_size may include the scalar second output)
  size_t tail = n4 * 4 + i;
  if (tail < n) out[tail] = 0.f;
}

// ---------------------------------------------------------------------------
// Kernel 2: one workgroup per batch sample.
//   Phase A: stream x[t, b, :] rows through a double-buffered LDS staging
//            area with async global->LDS copies; append events (t<<6|c).
//   Phase B: rank-sort events (unique packed keys => sort by t, then c,
//            identical to the reference's stable argsort order).
//   Phase C: cluster boundaries where time gap > WINDOW; mark clusters of
//            size exactly 2; scatter 1.0 into out[b, c, t].
// ---------------------------------------------------------------------------
__global__ __launch_bounds__(kThreads) void SCW_events_kernel(
    const float* __restrict__ x, float* __restrict__ out, int B) {
  const int b   = blockIdx.x;
  const int tid = threadIdx.x;
  const int rs  = B * kC;                // floats between consecutive t rows

  __shared__ __align__(16) float s_stage[2][kChunk * kC];  // 32 KiB
  __shared__ int           s_ev[kMaxEv];                   //  8 KiB
  __shared__ int           s_sorted[kMaxEv];               //  8 KiB
  __shared__ unsigned char s_bnd[kMaxEv + 1];              //  2 KiB
  __shared__ unsigned char s_mark[kMaxEv];                 //  2 KiB
  __shared__ int           s_n;

  if (tid == 0) s_n = 0;
  __syncthreads();

  const float* xb = x + (size_t)b * kC;

  // Stage one 16 KiB chunk (64 time rows x 64 ch) into LDS buffer `s`.
  // Each thread moves one float4; per wave this is ONE async instruction.
  auto issue = [&](int k, int s) {
    const int f4 = tid;                  // f4-per-chunk == kThreads
    const int r  = f4 >> 4;              // time row within chunk
    const int c4 = f4 & 15;              // float4 column
    const float* g = xb + (size_t)(k * kChunk + r) * rs + c4 * 4;
    float* l = &s_stage[s][f4 * 4];
#if SCW_ASYNC_LDS
    __builtin_amdgcn_global_load_async_to_lds_b128(
        (SCW_AS1 scw_v4i*)(void*)g, (SCW_AS3 scw_v4i*)(void*)l, 0, 0);
#else
    float4 v = *reinterpret_cast<const float4*>(g);
    *reinterpret_cast<float4*>(l) = v;
#endif
  };

  // -------- Phase A: double-buffered scan --------
  issue(0, 0);
  for (int k = 0; k < kNChunks; ++k) {
    const int s = k & 1;
    if (k + 1 < kNChunks) {
      issue(k + 1, s ^ 1);               // prefetch next chunk
      SCW_WAIT_ASYNC(1);                 // chunk k's copy (in-order) is done
    } else {
      SCW_WAIT_ASYNC(0);
    }
    __syncthreads();                     // all waves' copies for chunk k done

    {
      const int f4 = tid;
      float4 v = reinterpret_cast<const float4*>(s_stage[s])[f4];
      if (v.x != 0.f || v.y != 0.f || v.z != 0.f || v.w != 0.f) {
        const int t  = k * kChunk + (f4 >> 4);
        const int cb = (f4 & 15) * 4;
        const float vv[4] = {v.x, v.y, v.z, v.w};
#pragma unroll
        for (int j = 0; j < 4; ++j) {
          if (vv[j] != 0.f) {
            int slot = atomicAdd(&s_n, 1);           // ds_add_rtn_u32
            if (slot < kMaxEv) s_ev[slot] = (t << 6) | (cb + j);
          }
        }
      }
    }
    __syncthreads();                     // buffer s free for reuse at k+2
  }

  int n = s_n;
  if (n > kMaxEv) n = kMaxEv;

  // -------- Phase B: rank sort (keys unique => exact ranks) --------
  for (int i = tid; i < n; i += kThreads) {
    const int key = s_ev[i];
    int rank = 0;
    for (int j = 0; j < n; ++j) rank += (s_ev[j] < key) ? 1 : 0;
    s_sorted[rank] = key;
  }
  for (int i = tid; i < n; i += kThreads) s_mark[i] = 0;
  __syncthreads();

  // -------- Phase C: boundaries + size-2 cluster marking --------
  for (int i = tid; i < n; i += kThreads) {
    const bool bd =
        (i == 0) ||
        (((s_sorted[i] >> 6) - (s_sorted[i - 1] >> 6)) > kWindow);
    s_bnd[i] = bd ? 1 : 0;
  }
  if (tid == 0) s_bnd[n] = 1;            // sentinel: stream end closes cluster
  __syncthreads();

  // cluster starting at i has size exactly 2  <=>  bnd[i] & !bnd[i+1] & bnd[i+2]
  for (int i = tid; i + 2 <= n; i += kThreads) {
    if (s_bnd[i] && !s_bnd[i + 1] && s_bnd[i + 2]) {
      s_mark[i]     = 1;
      s_mark[i + 1] = 1;
    }
  }
  __syncthreads();

  // scatter 1.0 at marked events: out[b, c, t]
  for (int i = tid; i < n; i += kThreads) {
    if (s_mark[i]) {
      const int v = s_sorted[i];
      const int t = v >> 6;
      const int c = v & 63;
      out[((size_t)b * kC + c) * kT + t] = 1.0f;
    }
  }
}

// ---------------------------------------------------------------------------
extern "C" void kernel_launch(void* const* d_in, const int* in_sizes, int n_in,
                              void* d_out, int out_size, void* d_ws,
                              size_t ws_size, hipStream_t stream) {
  (void)n_in; (void)d_ws; (void)ws_size;
  const float* x = (const float*)d_in[0];
  float* out = (float*)d_out;

  const int B = in_sizes[0] / (kT * kC);   // 64

  // 1) zero full output (ordering guaranteed by same-stream serialization)
  SCW_zero_kernel<<<4096, 256, 0, stream>>>(out, (size_t)out_size);

  // 2) per-sample event detection / sort / cluster / scatter
  SCW_events_kernel<<<B, kThreads, 0, stream>>>(x, out, B);
}